// FSEncoder_5806795784467
// MI455X (gfx1250) — compile-verified
//
#include <hip/hip_runtime.h>

// Problem constants (match the reference)
#define BB     32
#define CINC   3
#define NN     4096
#define DIM    256
#define OUTD   256
#define PIECES 20

#define NTILE  64      // columns (points) per workgroup in the GEMM kernel
#define HPAD   258     // padded row stride of the LDS h1 tile (bank-conflict free b64 loads)

typedef float v2f __attribute__((ext_vector_type(2)));
typedef float v8f __attribute__((ext_vector_type(8)));

// ---------------------------------------------------------------------------
// Kernel 1: fused  h2 = W2 @ relu(W1 @ x + b1) + b2   (per batch GEMM, WMMA f32)
//   grid (NN/NTILE, BB), block 256 (8 wave32)
//   Each WG: batch b, columns [n0, n0+64), all 256 output rows.
//   Wave w computes rows [32w, 32w+32) x 64 cols as 2x4 tiles of 16x16.
// ---------------------------------------------------------------------------
__global__ __launch_bounds__(256) void k_conv_gemm(
    const float* __restrict__ x,  const float* __restrict__ W1,
    const float* __restrict__ b1, const float* __restrict__ W2,
    const float* __restrict__ b2, float* __restrict__ h2)
{
    __shared__ float hs1[NTILE][HPAD];   // h1 tile, [n][k] layout, k contiguous

    const int b   = blockIdx.y;
    const int n0  = blockIdx.x * NTILE;
    const int tid = threadIdx.x;

    // ---- stage 1: h1[n][k] = relu(W1[k,:] . x[b,:,n] + b1[k]) -------------
    {
        const int j     = tid & 63;        // column within tile
        const int kbase = (tid >> 6) * 64; // 4 groups of threads cover k in 64-chunks
        const float x0 = x[((size_t)b * CINC + 0) * NN + n0 + j];
        const float x1 = x[((size_t)b * CINC + 1) * NN + n0 + j];
        const float x2 = x[((size_t)b * CINC + 2) * NN + n0 + j];
        #pragma unroll 4
        for (int kk = 0; kk < 64; ++kk) {
            const int k = kbase + kk;
            float h = W1[k * 3 + 0] * x0 + W1[k * 3 + 1] * x1 + W1[k * 3 + 2] * x2 + b1[k];
            hs1[j][k] = fmaxf(h, 0.0f);
        }
    }
    __syncthreads();

    // ---- stage 2: WMMA GEMM over k = 0..255 step 4 ------------------------
    const int wave = tid >> 5;
    const int lane = tid & 31;
    const int lo   = lane & 15;
    const int hi   = lane >> 4;        // 0: lanes 0-15, 1: lanes 16-31
    const int cb   = wave * 32;        // first output row of this wave

    v8f acc[2][4];
    #pragma unroll
    for (int s = 0; s < 2; ++s)
        #pragma unroll
        for (int t = 0; t < 4; ++t)
            acc[s][t] = (v8f){0.f, 0.f, 0.f, 0.f, 0.f, 0.f, 0.f, 0.f};

    for (int k = 0; k < DIM; k += 4) {
        const int kk = k + 2 * hi;     // ISA A/B layout: hi half-wave holds K=2,3
        // A fragments (16x4 fp32): lane holds W2[m][kk], W2[m][kk+1]
        const v2f a0 = *(const v2f*)(W2 + (size_t)(cb + lo) * DIM + kk);
        const v2f a1 = *(const v2f*)(W2 + (size_t)(cb + 16 + lo) * DIM + kk);
        #pragma unroll
        for (int nt = 0; nt < 4; ++nt) {
            // B fragment (4x16 fp32): lane holds h1[kk][n], h1[kk+1][n]
            const v2f bf = *(const v2f*)(&hs1[nt * 16 + lo][kk]);
            acc[0][nt] = __builtin_amdgcn_wmma_f32_16x16x4_f32(
                false, a0, false, bf, (short)0, acc[0][nt], false, false);
            acc[1][nt] = __builtin_amdgcn_wmma_f32_16x16x4_f32(
                false, a1, false, bf, (short)0, acc[1][nt], false, false);
        }
    }

    // ---- stage 3: bias + store (C/D layout: VGPR r -> rows r / r+8) -------
    #pragma unroll
    for (int s = 0; s < 2; ++s)
        #pragma unroll
        for (int nt = 0; nt < 4; ++nt)
            #pragma unroll
            for (int r = 0; r < 8; ++r) {
                const int row = cb + s * 16 + r + 8 * hi;
                const int col = n0 + nt * 16 + lo;
                h2[((size_t)(b * DIM + row)) * NN + col] = acc[s][nt][r] + b2[row];
            }
}

// ---------------------------------------------------------------------------
// Kernel 2: FSPool — per (b,c): mask, stable descending argsort (bitonic on
// 64-bit keys in LDS, all lanes active: pair-indexed compare-exchange),
// perm output, piecewise-linear weighted sum -> pooled.
//   grid (DIM, BB), block 256
// ---------------------------------------------------------------------------
__global__ __launch_bounds__(256) void k_fspool(
    const float* __restrict__ h2, const int* __restrict__ n_points,
    const float* __restrict__ pool_w, int* __restrict__ perm,
    float* __restrict__ pooled)
{
    __shared__ unsigned long long kv[NN];   // 32 KB
    __shared__ float red[256];

    const int c   = blockIdx.x;
    const int b   = blockIdx.y;
    const int tid = threadIdx.x;

    const float denom = fmaxf((float)n_points[b] - 1.0f, 1.0f);
    const size_t base = ((size_t)(b * DIM + c)) * NN;

    // Build sortable keys: ascending sort of (~ordered(hm) << 32 | idx)
    //  == descending by value, ascending index tie-break (stable argsort(-hm)).
    for (int j = tid; j < NN; j += 256) {
        float v = h2[base + j];
        if ((float)j > denom) v -= 99999.0f;            // (1-mask)*(-99999)
        unsigned u = __float_as_uint(v);
        u = (u & 0x80000000u) ? ~u : (u | 0x80000000u); // monotonic ascending
        u = ~u;                                          // flip for descending
        kv[j] = ((unsigned long long)u << 32) | (unsigned)j;
    }

    // Bitonic sort, ascending over combined key.
    // Pair-indexed: 2048 compare-exchanges per pass, every lane active.
    for (int k = 2; k <= NN; k <<= 1) {
        for (int j = k >> 1; j > 0; j >>= 1) {
            __syncthreads();
            #pragma unroll 2
            for (int t = tid; t < (NN / 2); t += 256) {
                const int i   = 2 * t - (t & (j - 1));   // bit log2(j) of i is 0
                const int ixj = i + j;                   // == i ^ j
                const unsigned long long a0 = kv[i];
                const unsigned long long a1 = kv[ixj];
                const bool up = ((i & k) == 0);
                if ((a0 > a1) == up) { kv[i] = a1; kv[ixj] = a0; }
            }
        }
    }
    __syncthreads();

    // Emit perm, decode values, apply rank mask + piecewise-linear weights.
    float accv = 0.0f;
    for (int j = tid; j < NN; j += 256) {
        const unsigned long long e = kv[j];
        const int idx = (int)(unsigned)(e & 0xFFFFFFFFull);
        perm[base + j] = idx;

        unsigned u = ~(unsigned)(e >> 32);               // undo descending flip
        const unsigned bits = (u & 0x80000000u) ? (u ^ 0x80000000u) : ~u;
        const float hv = __uint_as_float(bits);

        const float fj    = (float)j;
        const float maskr = (fj <= denom) ? 1.0f : 0.0f;
        const float ratio = fminf(fj / denom, 1.0f);
        const float fi    = (float)PIECES * ratio;
        int   ii   = (int)fi; if (ii > PIECES) ii = PIECES;
        const float frac = fi - (float)ii;
        int   ir   = ii + 1; if (ir > PIECES) ir = PIECES;
        const float w = (1.0f - frac) * pool_w[c * (PIECES + 1) + ii]
                      +         frac  * pool_w[c * (PIECES + 1) + ir];
        accv += hv * w * maskr;
    }

    red[tid] = accv;
    __syncthreads();
    for (int s = 128; s > 0; s >>= 1) {
        if (tid < s) red[tid] += red[tid + s];
        __syncthreads();
    }
    if (tid == 0) pooled[b * DIM + c] = red[0];
}

// ---------------------------------------------------------------------------
// Kernel 3: tiny head  y = relu(pooled @ L1^T + bl1) @ L2^T + bl2
//   grid BB, block 256 (one thread per output feature)
// ---------------------------------------------------------------------------
__global__ __launch_bounds__(256) void k_mlp(
    const float* __restrict__ pooled,
    const float* __restrict__ L1, const float* __restrict__ bl1,
    const float* __restrict__ L2, const float* __restrict__ bl2,
    float* __restrict__ y)
{
    __shared__ float pb[DIM];
    __shared__ float t1[DIM];
    const int b   = blockIdx.x;
    const int tid = threadIdx.x;

    pb[tid] = pooled[b * DIM + tid];
    __syncthreads();

    float s = bl1[tid];
    #pragma unroll 8
    for (int k = 0; k < DIM; ++k) s += L1[tid * DIM + k] * pb[k];
    t1[tid] = fmaxf(s, 0.0f);
    __syncthreads();

    float s2 = bl2[tid];
    #pragma unroll 8
    for (int k = 0; k < DIM; ++k) s2 += L2[tid * DIM + k] * t1[k];
    y[b * OUTD + tid] = s2;
}

// ---------------------------------------------------------------------------
extern "C" void kernel_launch(void* const* d_in, const int* in_sizes, int n_in,
                              void* d_out, int out_size, void* d_ws, size_t ws_size,
                              hipStream_t stream)
{
    const float* x        = (const float*)d_in[0];
    const int*   n_points = (const int*)  d_in[1];
    const float* W1       = (const float*)d_in[2];
    const float* b1       = (const float*)d_in[3];
    const float* W2       = (const float*)d_in[4];
    const float* b2       = (const float*)d_in[5];
    const float* pool_w   = (const float*)d_in[6];
    const float* L1       = (const float*)d_in[7];
    const float* bl1      = (const float*)d_in[8];
    const float* L2       = (const float*)d_in[9];
    const float* bl2      = (const float*)d_in[10];

    // Workspace: h2 (32*256*4096 f32 = 128 MB) + pooled (32*256 f32)
    float* h2     = (float*)d_ws;
    float* pooled = (float*)((char*)d_ws + (size_t)BB * DIM * NN * sizeof(float));

    // Output: y (32*256 f32) followed by perm (32*256*4096 i32), flat.
    float* y    = (float*)d_out;
    int*   perm = (int*)d_out + BB * OUTD;

    dim3 g1(NN / NTILE, BB);
    k_conv_gemm<<<g1, 256, 0, stream>>>(x, W1, b1, W2, b2, h2);

    dim3 g2(DIM, BB);
    k_fspool<<<g2, 256, 0, stream>>>(h2, n_points, pool_w, perm, pooled);

    k_mlp<<<BB, 256, 0, stream>>>(pooled, L1, bl1, L2, bl2, y);
}